// HierarchicalIntentClassifier_42906723287631
// MI455X (gfx1250) — compile-verified
//
#include <hip/hip_runtime.h>
#include <hip/hip_bf16.h>

#define B_N 8192
#define D_N 768
#define E_N 8
#define K_N 32

typedef __attribute__((ext_vector_type(2))) float v2f;
typedef __attribute__((ext_vector_type(8))) float v8f;
typedef __attribute__((ext_vector_type(4))) unsigned int u32x4;
typedef __attribute__((ext_vector_type(8))) unsigned int u32x8;

// ---------------------------------------------------------------------------
// Kernel 1: main_out = pooled @ main_W + main_b  (V_WMMA_F32_16X16X4_F32),
//           per-row argmax (first-max, matches jnp.argmax), expert binning.
// 128 threads = 4 waves; each wave owns one 16-row output tile.
// B staged zero-padded (8->16 cols) and pair-interleaved so the K-loop's B
// operand is one unconditional ds_load_b64 (no exec-masked LDS loads).
// ---------------------------------------------------------------------------
__global__ __launch_bounds__(128) void k_main_route(
    const float* __restrict__ pooled, const float* __restrict__ main_W,
    const float* __restrict__ main_b, float* __restrict__ main_out,
    int* __restrict__ counts, int* __restrict__ lists)
{
    __shared__ v2f   sB[(D_N / 2) * 16];   // 48 KB: (B[2p][m], B[2p+1][m])
    __shared__ float sTile[4][16][E_N];    // 2 KB: logits for argmax

    const int tid = threadIdx.x;
    for (int i = tid; i < (D_N / 2) * 16; i += 128) {
        const int p = i >> 4, col = i & 15;
        v2f v; v.x = 0.f; v.y = 0.f;
        if (col < E_N) {
            v.x = main_W[(2 * p)     * E_N + col];
            v.y = main_W[(2 * p + 1) * E_N + col];
        }
        sB[i] = v;
    }
    __syncthreads();

    const int wave = tid >> 5;
    const int lane = tid & 31;
    const int m    = lane & 15;            // A: row M=m ; B/C/D: column N=m
    const int h    = lane >> 4;            // half-wave: K=2h..2h+1 / M+8h
    const long rowBase = ((long)blockIdx.x * 4 + wave) * 16;

    const float* aPtr = pooled + (rowBase + m) * D_N;
    v8f c = {};
#pragma unroll 8
    for (int k = 0; k < D_N; k += 4) {
        v2f a;
        // A 16x4 f32: vgpr0 = K=0|2, vgpr1 = K=1|3 (lanes 0-15 | 16-31)
        a.x = aPtr[k + 2 * h];
        a.y = aPtr[k + 2 * h + 1];
        // B 4x16 f32: rows (k+2h, k+2h+1) at column m, one b64 LDS load
        const v2f b = sB[((k >> 1) + h) * 16 + m];
        c = __builtin_amdgcn_wmma_f32_16x16x4_f32(
                false, a, false, b, (short)0, c, false, false);
    }

    // C/D layout: lane m = column, VGPR r = row (r + 8h)
    if (m < E_N) {
        const float bias = main_b[m];
        for (int r = 0; r < 8; ++r) {
            const int rloc = r + 8 * h;
            const float v = c[r] + bias;
            main_out[(rowBase + rloc) * E_N + m] = v;
            sTile[wave][rloc][m] = v;
        }
    }
    __syncthreads();

    // 64 rows per block; one thread per row does argmax + binning
    if (tid < 64) {
        const int rowg = blockIdx.x * 64 + tid;
        const int w2 = tid >> 4, rl = tid & 15;
        float best = sTile[w2][rl][0];
        int bi = 0;
        for (int e = 1; e < E_N; ++e) {
            const float v = sTile[w2][rl][e];
            if (v > best) { best = v; bi = e; }   // strict > keeps first max
        }
        const int pos = atomicAdd(&counts[bi], 1);
        lists[bi * B_N + pos] = rowg;
    }
}

// ---------------------------------------------------------------------------
// Kernel 2: grouped GEMM over the expert bins.
// Block (e, tg): TDM (tensor_load_to_lds) stages sub_W[e] (96 KB) into LDS,
// issued by wave 0 and tracked with TENSORcnt; 8 waves each compute a
// gathered 16x32 tile with two f32 WMMA accumulators; bias + size-mask.
// ---------------------------------------------------------------------------
__global__ __launch_bounds__(256) void k_sub_expert(
    const float* __restrict__ pooled, const float* __restrict__ sub_W,
    const float* __restrict__ sub_b, const int* __restrict__ sub_sizes,
    const int* __restrict__ counts, const int* __restrict__ lists,
    float* __restrict__ sub_out)
{
    __shared__ float sW[D_N * K_N];        // 96 KB (CDNA5 WGP LDS = 320 KB)

    const int e  = blockIdx.x >> 6;        // expert id
    const int tg = blockIdx.x & 63;        // 128-row group within expert
    const int cnt = counts[e];
    if (tg * 128 >= cnt) return;           // block-uniform early-out

    const int tid = threadIdx.x;

    // --- TDM stage: one descriptor copies 24576 contiguous f32 -> LDS ------
    if (tid < 32) {                        // wave 0 only issues the DMA
        const unsigned long long ga =
            (unsigned long long)(uintptr_t)(sub_W + (size_t)e * D_N * K_N);
        const unsigned int ldsa = (unsigned int)(uintptr_t)(void*)sW; // LDS byte addr
        // D# group 0: count=1 | lds_addr | global_addr[56:0] | type=2
        const u32x4 g0 = { 1u, ldsa, (unsigned int)ga,
                           ((unsigned int)(ga >> 32) & 0x01FFFFFFu) | 0x80000000u };
        // D# group 1: data_size=4B; dim0=stride0=tile0=24576; dim1=tile1=1
        const u32x8 g1 = { 0x00020000u,    // mask=0, data_size=2 (4B)
                           0x60000000u,    // barrier_addr=0, tensor_dim0.lo=24576
                           0x00010000u,    // tensor_dim0.hi=0, tensor_dim1=1
                           0x60000000u,    // tensor_dim1.hi=0, tile_dim0=24576
                           0x00000001u,    // tile_dim1=1, tile_dim2=0
                           24576u,         // tensor_dim0_stride.lo
                           0x60000000u,    // stride0.hi=0, tensor_dim1_stride.lo
                           0u };           // tensor_dim1_stride.hi
        asm volatile("tensor_load_to_lds %0, %1" :: "s"(g0), "s"(g1) : "memory");
        __builtin_amdgcn_s_wait_tensorcnt(0);
    }
    __syncthreads();

    const int wave = tid >> 5;
    const int lane = tid & 31;
    const int m    = lane & 15;
    const int h    = lane >> 4;
    const int base = tg * 128 + wave * 16;
    if (base >= cnt) return;               // wave-uniform, after the barrier

    const int* myList = lists + e * B_N;
    const int p   = base + m;
    const int row = myList[(p < cnt) ? p : base];   // clamp padding rows

    const float* aPtr = pooled + (size_t)row * D_N;
    v8f c0 = {}, c1 = {};
#pragma unroll 4
    for (int k = 0; k < D_N; k += 4) {
        v2f a, b0, b1;
        a.x = aPtr[k + 2 * h];
        a.y = aPtr[k + 2 * h + 1];
        const int r0 = (k + 2 * h) * K_N;
        const int r1 = r0 + K_N;
        b0.x = sW[r0 + m];       b0.y = sW[r1 + m];        // columns 0..15
        b1.x = sW[r0 + 16 + m];  b1.y = sW[r1 + 16 + m];   // columns 16..31
        c0 = __builtin_amdgcn_wmma_f32_16x16x4_f32(
                false, a, false, b0, (short)0, c0, false, false);
        c1 = __builtin_amdgcn_wmma_f32_16x16x4_f32(
                false, a, false, b1, (short)0, c1, false, false);
    }

    const int   ssz   = sub_sizes[e];
    const float bias0 = sub_b[e * K_N + m];
    const float bias1 = sub_b[e * K_N + 16 + m];
    const float keep0 = (m < ssz)        ? 1.f : 0.f;
    const float keep1 = ((16 + m) < ssz) ? 1.f : 0.f;
    for (int r = 0; r < 8; ++r) {
        const int pr = base + r + 8 * h;
        if (pr < cnt) {
            const int rid = myList[pr];
            float* o = sub_out + (size_t)rid * K_N;
            o[m]      = (c0[r] + bias0) * keep0;
            o[16 + m] = (c1[r] + bias1) * keep1;
        }
    }
}

// ---------------------------------------------------------------------------
extern "C" void kernel_launch(void* const* d_in, const int* in_sizes, int n_in,
                              void* d_out, int out_size, void* d_ws, size_t ws_size,
                              hipStream_t stream) {
    const float* pooled    = (const float*)d_in[0];
    const float* main_W    = (const float*)d_in[1];
    const float* main_b    = (const float*)d_in[2];
    const float* sub_W     = (const float*)d_in[3];
    const float* sub_b     = (const float*)d_in[4];
    const int*   sub_sizes = (const int*)d_in[5];

    float* main_out = (float*)d_out;                 // [B, E]
    float* sub_out  = (float*)d_out + B_N * E_N;     // [B, K]

    int* counts = (int*)d_ws;                        // E counters (padded to 16)
    int* lists  = (int*)d_ws + 16;                   // E * B row ids

    hipMemsetAsync(counts, 0, 16 * sizeof(int), stream);

    k_main_route<<<B_N / 64, 128, 0, stream>>>(
        pooled, main_W, main_b, main_out, counts, lists);

    k_sub_expert<<<E_N * 64, 256, 0, stream>>>(
        pooled, sub_W, sub_b, sub_sizes, counts, lists, sub_out);
}